// Linear_49074296324503
// MI455X (gfx1250) — compile-verified
//
#include <hip/hip_runtime.h>
#include <hip/hip_bf16.h>

// Reference constants (from the Python reference): BLOCK=32, HO=WO=128,
// N_TOK=8192, in_features = out_features = 4096.
#define BLOCK   32
#define HO_DIM  128
#define FEATS   4096
#define NTOK    8192

typedef float v2f __attribute__((ext_vector_type(2)));
typedef float v8f __attribute__((ext_vector_type(8)));

// ---------------------------------------------------------------------------
// Kernel A: build a tiny CSR over the block LUT in workspace.
//   row_ptr[HO_DIM+1], then pairs[nnz] = (block_index k, block_col ci[k]).
// One workgroup of HO_DIM threads; thread r owns output block-row r.
// ---------------------------------------------------------------------------
__global__ void build_csr_kernel(const int* __restrict__ ri,
                                 const int* __restrict__ ci,
                                 int nnz,
                                 int* __restrict__ row_ptr,
                                 int2* __restrict__ pairs) {
    __shared__ int cnt[HO_DIM];
    __shared__ int off[HO_DIM];
    const int r = threadIdx.x;   // blockDim.x == HO_DIM

    int c = 0;
    for (int k = 0; k < nnz; ++k) c += (ri[k] == r) ? 1 : 0;
    cnt[r] = c;
    __syncthreads();

    if (r == 0) {
        int s = 0;
        for (int i = 0; i < HO_DIM; ++i) { off[i] = s; s += cnt[i]; }
    }
    __syncthreads();

    int o = off[r];
    row_ptr[r] = o;
    if (r == HO_DIM - 1) row_ptr[HO_DIM] = o + cnt[r];

    for (int k = 0; k < nnz; ++k) {
        if (ri[k] == r) {
            pairs[o] = make_int2(k, ci[k]);
            ++o;
        }
    }
}

// ---------------------------------------------------------------------------
// Kernel B: block-sparse GEMM on the f32 WMMA pipe.
//   grid.x = HO_DIM (output block-row), grid.y = NTOK / 128 (token tiles)
//   block  = 128 threads = 4 waves; each wave: 32 tokens x 32 outputs.
// ---------------------------------------------------------------------------
__global__ void __launch_bounds__(128)
dsd_wmma_f32_kernel(const float* __restrict__ x,
                    const float* __restrict__ w_blocks,
                    const int*  __restrict__ row_ptr,
                    const int2* __restrict__ pairs,
                    float* __restrict__ y) {
    const int r    = blockIdx.x;             // output block-row (0..127)
    const int wave = threadIdx.x >> 5;       // 0..3
    const int lane = threadIdx.x & 31;
    const int mrow = lane & 15;              // M (and B-column) index 0..15
    const int hi   = lane >> 4;              // 0: K={0,1}, 1: K={2,3} half

    const int t0 = blockIdx.y * 128 + wave * 32;   // first token of this wave

    // acc[ts*2 + nh]: ts = token half (16 tokens), nh = output half (16 cols)
    v8f acc0 = {}, acc1 = {}, acc2 = {}, acc3 = {};

    const int kb = row_ptr[r];
    const int ke = row_ptr[r + 1];

    // x rows for this lane's M index, for both 16-token sets
    const float* xrow0 = x + (size_t)(t0 + mrow) * FEATS;
    const float* xrow1 = xrow0 + (size_t)16 * FEATS;

    for (int p = kb; p < ke; ++p) {
        const int2 pr = pairs[p];                       // uniform -> s_load
        const float* wb = w_blocks + (size_t)pr.x * (BLOCK * BLOCK); // [a][b] row-major
        const int cbase = pr.y * BLOCK;

        const float* xr0 = xrow0 + cbase;
        const float* xr1 = xrow1 + cbase;

#pragma unroll
        for (int kk = 0; kk < BLOCK; kk += 4) {
            // A (16x4 f32): lane(mrow,hi) holds K = kk + hi*2 and +1
            v2f a0 = *(const v2f*)(xr0 + kk + hi * 2);
            v2f a1 = *(const v2f*)(xr1 + kk + hi * 2);
            // B (4x16 f32): B[k][n] = w[n][k]; lane column n = mrow,
            // rows K = kk + hi*2 (+1). Output half 1 uses n+16.
            v2f b0 = *(const v2f*)(wb + (size_t)mrow * BLOCK        + kk + hi * 2);
            v2f b1 = *(const v2f*)(wb + (size_t)(mrow + 16) * BLOCK + kk + hi * 2);

            acc0 = __builtin_amdgcn_wmma_f32_16x16x4_f32(false, a0, false, b0,
                                                         (short)0, acc0, false, false);
            acc1 = __builtin_amdgcn_wmma_f32_16x16x4_f32(false, a0, false, b1,
                                                         (short)0, acc1, false, false);
            acc2 = __builtin_amdgcn_wmma_f32_16x16x4_f32(false, a1, false, b0,
                                                         (short)0, acc2, false, false);
            acc3 = __builtin_amdgcn_wmma_f32_16x16x4_f32(false, a1, false, b1,
                                                         (short)0, acc3, false, false);
        }
    }

    // D layout (16x16 f32 C/D): VGPR j, lane -> M = j + hi*8, N = mrow.
    const int col0 = r * BLOCK + mrow;
    {
        float* yr = y + (size_t)(t0 + hi * 8) * FEATS + col0;
#pragma unroll
        for (int j = 0; j < 8; ++j) {
            yr[(size_t)j * FEATS]      = acc0[j];
            yr[(size_t)j * FEATS + 16] = acc1[j];
        }
    }
    {
        float* yr = y + (size_t)(t0 + 16 + hi * 8) * FEATS + col0;
#pragma unroll
        for (int j = 0; j < 8; ++j) {
            yr[(size_t)j * FEATS]      = acc2[j];
            yr[(size_t)j * FEATS + 16] = acc3[j];
        }
    }
}

// ---------------------------------------------------------------------------
// Launch wrapper
// ---------------------------------------------------------------------------
extern "C" void kernel_launch(void* const* d_in, const int* in_sizes, int n_in,
                              void* d_out, int out_size, void* d_ws, size_t ws_size,
                              hipStream_t stream) {
    const float* x        = (const float*)d_in[0];
    const float* w_blocks = (const float*)d_in[1];
    const int*   ri       = (const int*)d_in[2];
    const int*   ci       = (const int*)d_in[3];
    const int    nnz      = in_sizes[2];

    int*  row_ptr = (int*)d_ws;                          // (HO_DIM+1) ints
    int2* pairs   = (int2*)((char*)d_ws + 256);          // nnz int2, 256B aligned

    build_csr_kernel<<<1, HO_DIM, 0, stream>>>(ri, ci, nnz, row_ptr, pairs);

    dim3 grid(HO_DIM, NTOK / 128);
    dim3 block(128);
    dsd_wmma_f32_kernel<<<grid, block, 0, stream>>>(x, w_blocks, row_ptr, pairs,
                                                    (float*)d_out);
}